// GRUDecoder_33809982554324
// MI455X (gfx1250) — compile-verified
//
#include <hip/hip_runtime.h>

// ---------------------------------------------------------------------------
// GRU decoder step (Bahdanau attention + GRU cell + vocab projection), fp32.
// MI455X / gfx1250: wave32, WMMA f32 16x16x4 for the GEMMs, async global->LDS
// staging (ASYNCcnt) for the A tiles.
// Memory-bound: ~240 MB traffic (fc_W = 196.6 MB dominates) -> ~10 us floor.
// ---------------------------------------------------------------------------

typedef __attribute__((ext_vector_type(2))) float v2f;
typedef __attribute__((ext_vector_type(8))) float v8f;

static constexpr int kB   = 128;    // batch
static constexpr int kS   = 32;     // src len
static constexpr int kD   = 512;    // EMB == ENC == DEC
static constexpr int kV   = 32000;  // vocab
static constexpr int kFCK = 1536;   // DEC + ENC + EMB
static constexpr int kLDK = 516;    // 512 + 4 pad; 516*4 B = 2064 (16B aligned)

__device__ __forceinline__ v8f wmma_f32_16x16x4(v2f a, v2f b, v8f c) {
  // 8-arg pattern per probe: (neg_a, A, neg_b, B, c_mod, C, reuse_a, reuse_b)
  return __builtin_amdgcn_wmma_f32_16x16x4_f32(false, a, false, b, (short)0, c,
                                               false, false);
}

// Async DMA one 16-byte packet per lane from global (SGPR base + u32 byte
// offset) directly into LDS (u32 byte address). Tracked by ASYNCcnt.
__device__ __forceinline__ void async_g2l_b128(unsigned lds_byte_addr,
                                               unsigned glb_byte_off,
                                               const float* base) {
  asm volatile("global_load_async_to_lds_b128 %0, %1, %2"
               :
               : "v"(lds_byte_addr), "v"(glb_byte_off), "s"(base)
               : "memory");
}
__device__ __forceinline__ void wait_async0() {
  asm volatile("s_wait_asynccnt 0" ::: "memory");
}

// Stage a 16 x 512 fp32 tile (rows m_base.., cols kb..kb+511 of row-major A
// with row stride K floats) into padded LDS via async loads.
// 256 threads: thread t moves 16 B; 2 rows per pass, 8 passes.
__device__ __forceinline__ void stage_tile_async(float* lds, const float* A,
                                                 int m_base, int kb, int K,
                                                 int t) {
  const int r2 = t >> 7;          // 0..1 : row within pass
  const int cid = (t & 127) * 4;  // column (floats), multiples of 4
#pragma unroll
  for (int rr = 0; rr < 16; rr += 2) {
    const int r = rr + r2;
    const unsigned laddr =
        (unsigned)(uintptr_t)(&lds[r * kLDK + cid]);  // low 32 = LDS offset
    const unsigned goff =
        (unsigned)(((size_t)(m_base + r) * K + kb + cid) * sizeof(float));
    async_g2l_b128(laddr, goff, A);
  }
  wait_async0();   // this wave's DMA done (LDS writes landed)
  __syncthreads(); // all waves' tiles visible
}

// ---------------------------------------------------------------------------
// 1) Embedding gather into fcin[:, 1024:1536]
// ---------------------------------------------------------------------------
__global__ __launch_bounds__(256) void k_embed(const int* __restrict__ tok,
                                               const float* __restrict__ emb,
                                               float* __restrict__ fcin) {
  const int b = blockIdx.x;
  const int row = tok[b];
  for (int i = threadIdx.x; i < kD; i += 256)
    fcin[(size_t)b * kFCK + 1024 + i] = emb[(size_t)row * kD + i];
}

// ---------------------------------------------------------------------------
// 2) Generic fp32 WMMA GEMM:  C(MxN) = A(MxK) @ B(KxN) + bias[n]
//    A row-major (lda == K), B row-major (ldb == N).
//    Workgroup: 256 threads = 8 waves; wave computes a 16(M) x 32(N) tile.
//    grid = (N/256, M/16).  K must be a multiple of 512 (chunked via LDS).
//    A fragment (ISA 16x4 f32 layout): lane L -> M = L%16,
//      vgpr0 = A[M][k + 2*(L/16)], vgpr1 = A[M][k + 2*(L/16) + 1].
//    B fragment mirrored: vgpr0 = B[k + 2*(L/16)][n], vgpr1 = next K row.
// ---------------------------------------------------------------------------
__global__ __launch_bounds__(256) void k_gemm_bias(
    const float* __restrict__ A, const float* __restrict__ Bm,
    const float* __restrict__ bias, float* __restrict__ C, int N, int K) {
  __shared__ __attribute__((aligned(16))) float lds[16 * kLDK];
  const int t = threadIdx.x;
  const int lane = t & 31, wave = t >> 5;
  const int l15 = lane & 15;
  const int kk = (lane >> 4) * 2;
  const int m_base = blockIdx.y * 16;
  const int n0 = blockIdx.x * 256 + wave * 32;

  v8f c0 = {};
  v8f c1 = {};
  for (int kb = 0; kb < K; kb += 512) {
    stage_tile_async(lds, A, m_base, kb, K, t);

#pragma unroll 4
    for (int k = 0; k < 512; k += 4) {
      v2f a = {lds[l15 * kLDK + k + kk], lds[l15 * kLDK + k + kk + 1]};
      const float* Bk = Bm + (size_t)(kb + k + kk) * N;
      v2f b0 = {Bk[n0 + l15], Bk[(size_t)N + n0 + l15]};
      c0 = wmma_f32_16x16x4(a, b0, c0);
      v2f b1 = {Bk[16 + n0 + l15], Bk[(size_t)N + 16 + n0 + l15]};
      c1 = wmma_f32_16x16x4(a, b1, c1);
    }
    __syncthreads();
  }

  // C layout: lane L holds column n = n0 + L%16, rows m_base + 8*(L/16) + i.
  const int mrow = m_base + (lane >> 4) * 8;
  const float bi0 = bias[n0 + l15];
  const float bi1 = bias[n0 + 16 + l15];
#pragma unroll
  for (int i = 0; i < 8; ++i) {
    C[(size_t)(mrow + i) * N + n0 + l15] = c0[i] + bi0;
    C[(size_t)(mrow + i) * N + n0 + 16 + l15] = c1[i] + bi1;
  }
}

// ---------------------------------------------------------------------------
// 3) Attention energy:  rows r = b*32+s of enc(4096x512) @ attn_W[512:](512x512)
//    Fused epilogue: score[r] += sum_n v[n] * tanh(e[r,n] + hW[b,n])
//    (hW already includes attn_b).  Wave = one 16x16 tile; width-16 shuffle
//    reduction, one atomicAdd per row per tile.  grid = (512/128, 4096/16).
// ---------------------------------------------------------------------------
__global__ __launch_bounds__(256) void k_energy(
    const float* __restrict__ enc, const float* __restrict__ Bm,
    const float* __restrict__ hW, const float* __restrict__ v_w,
    float* __restrict__ scores) {
  __shared__ __attribute__((aligned(16))) float lds[16 * kLDK];
  const int t = threadIdx.x;
  const int lane = t & 31, wave = t >> 5;
  const int l15 = lane & 15;
  const int kk = (lane >> 4) * 2;
  const int m_base = blockIdx.y * 16;
  const int n0 = blockIdx.x * 128 + wave * 16;

  stage_tile_async(lds, enc, m_base, 0, kD, t);

  v8f c0 = {};
#pragma unroll 4
  for (int k = 0; k < 512; k += 4) {
    v2f a = {lds[l15 * kLDK + k + kk], lds[l15 * kLDK + k + kk + 1]};
    const float* Bk = Bm + (size_t)(k + kk) * kD;
    v2f b = {Bk[n0 + l15], Bk[kD + n0 + l15]};
    c0 = wmma_f32_16x16x4(a, b, c0);
  }

  const int n = n0 + l15;
  const float vn = v_w[n];
  const int mrow = m_base + (lane >> 4) * 8;
  float p[8];
#pragma unroll
  for (int i = 0; i < 8; ++i) {
    const int m = mrow + i;  // flattened b*32 + s
    const int bb = m >> 5;
    p[i] = vn * tanhf(c0[i] + hW[(size_t)bb * kD + n]);
  }
#pragma unroll
  for (int i = 0; i < 8; ++i) {
    float s = p[i];
    for (int off = 1; off < 16; off <<= 1) s += __shfl_xor(s, off, 16);
    if (l15 == 0) atomicAdd(&scores[mrow + i], s);
  }
}

// ---------------------------------------------------------------------------
// 4) Softmax over S=32 + context; attn weights -> d_out, context -> fcin.
// ---------------------------------------------------------------------------
__global__ __launch_bounds__(256) void k_softmax_ctx(
    const float* __restrict__ scores, const unsigned char* __restrict__ mask,
    const float* __restrict__ enc, float* __restrict__ attn_out,
    float* __restrict__ fcin) {
  __shared__ float sc[kS];
  const int b = blockIdx.x, t = threadIdx.x;
  if (t < kS) {
    float v = scores[b * kS + t];
    if (mask[b * kS + t]) v = -__builtin_huge_valf();
    sc[t] = v;
  }
  __syncthreads();

  float mx = -__builtin_huge_valf();
  for (int s = 0; s < kS; ++s) mx = fmaxf(mx, sc[s]);
  float wl[kS];
  float sum = 0.f;
  for (int s = 0; s < kS; ++s) {
    wl[s] = expf(sc[s] - mx);
    sum += wl[s];
  }
  const float inv = 1.f / sum;
  for (int s = 0; s < kS; ++s) wl[s] *= inv;

  if (t < kS) attn_out[b * kS + t] = wl[t];
  for (int nidx = t; nidx < kD; nidx += 256) {
    float acc = 0.f;
    for (int s = 0; s < kS; ++s)
      acc += wl[s] * enc[((size_t)b * kS + s) * kD + nidx];
    fcin[(size_t)b * kFCK + 512 + nidx] = acc;
  }
}

// ---------------------------------------------------------------------------
// 5) GRU cell (one WG per batch row).  Split-K dot products: lanes stride the
//    K dimension -> fully coalesced weight reads (W_ih/W_hh stay L2-resident).
//    x = [embedded | context]; gate order r, z, n (PyTorch).
// ---------------------------------------------------------------------------
__global__ __launch_bounds__(256) void k_gru(
    float* __restrict__ fcin, const float* __restrict__ hidden,
    const float* __restrict__ W_ih, const float* __restrict__ W_hh,
    const float* __restrict__ b_ih, const float* __restrict__ b_hh,
    float* __restrict__ out_h) {
  __shared__ float xl[1024];
  __shared__ float hl[kD];
  __shared__ float rz[1024];
  __shared__ float gxn[kD];
  __shared__ float ghn[kD];
  const int b = blockIdx.x, t = threadIdx.x;
  const int lane = t & 31, wave = t >> 5;

  for (int i = t; i < kD; i += 256) {
    xl[i] = fcin[(size_t)b * kFCK + 1024 + i];       // embedded
    xl[512 + i] = fcin[(size_t)b * kFCK + 512 + i];  // context
    hl[i] = hidden[(size_t)b * kD + i];
  }
  __syncthreads();

  for (int j = wave; j < 3 * kD; j += 8) {
    const float* wi = W_ih + (size_t)j * 1024;
    const float* wh = W_hh + (size_t)j * kD;
    float gx = 0.f, gh = 0.f;
    for (int k = lane; k < 1024; k += 32) gx += xl[k] * wi[k];
    for (int k = lane; k < kD; k += 32) gh += hl[k] * wh[k];
#pragma unroll
    for (int off = 16; off > 0; off >>= 1) {
      gx += __shfl_xor(gx, off, 32);
      gh += __shfl_xor(gh, off, 32);
    }
    if (lane == 0) {
      gx += b_ih[j];
      gh += b_hh[j];
      if (j < 1024) rz[j] = gx + gh;
      else {
        gxn[j - 1024] = gx;
        ghn[j - 1024] = gh;
      }
    }
  }
  __syncthreads();

  for (int d = t; d < kD; d += 256) {
    const float r = 1.f / (1.f + expf(-rz[d]));
    const float z = 1.f / (1.f + expf(-rz[512 + d]));
    const float n = tanhf(gxn[d] + r * ghn[d]);
    const float hn = (1.f - z) * n + z * hl[d];
    fcin[(size_t)b * kFCK + d] = hn;  // for fc input
    out_h[(size_t)b * kD + d] = hn;   // hidden output
  }
}

// ---------------------------------------------------------------------------
// launch
// ---------------------------------------------------------------------------
extern "C" void kernel_launch(void* const* d_in, const int* in_sizes, int n_in,
                              void* d_out, int out_size, void* d_ws,
                              size_t ws_size, hipStream_t stream) {
  const int* tok = (const int*)d_in[0];
  const float* hidden = (const float*)d_in[1];       // (1,B,512)
  const float* enc = (const float*)d_in[2];          // (B,S,512)
  const unsigned char* mask = (const unsigned char*)d_in[3];
  const float* emb = (const float*)d_in[4];          // (V,512)
  const float* attn_W = (const float*)d_in[5];       // (1024,512)
  const float* attn_b = (const float*)d_in[6];       // (512)
  const float* v_w = (const float*)d_in[7];          // (512)
  const float* W_ih = (const float*)d_in[8];         // (1536,1024)
  const float* W_hh = (const float*)d_in[9];         // (1536,512)
  const float* b_ih = (const float*)d_in[10];
  const float* b_hh = (const float*)d_in[11];
  const float* fc_W = (const float*)d_in[12];        // (1536,32000)
  const float* fc_b = (const float*)d_in[13];

  // d_out: prediction (B,V) | h_new (1,B,512) | attn_weights (B,S)
  float* pred = (float*)d_out;
  float* out_h = pred + (size_t)kB * kV;
  float* attn_out = out_h + (size_t)kB * kD;

  // workspace: hW (B,512) | scores (B,S) | fcin (B,1536)
  float* ws = (float*)d_ws;
  float* hW = ws;
  float* scores = hW + (size_t)kB * kD;
  float* fcin = scores + (size_t)kB * kS;

  k_embed<<<dim3(kB), dim3(256), 0, stream>>>(tok, emb, fcin);

  // hW = h_last @ attn_W[0:512,:] + attn_b   (M=128,N=512,K=512)
  k_gemm_bias<<<dim3(kD / 256, kB / 16), dim3(256), 0, stream>>>(
      hidden, attn_W, attn_b, hW, kD, kD);

  hipMemsetAsync(scores, 0, (size_t)kB * kS * sizeof(float), stream);

  // scores += sum_n v[n]*tanh(enc@attn_W[512:] + hW)   (M=4096,N=512,K=512)
  k_energy<<<dim3(kD / 128, (kB * kS) / 16), dim3(256), 0, stream>>>(
      enc, attn_W + (size_t)kD * kD, hW, v_w, scores);

  k_softmax_ctx<<<dim3(kB), dim3(256), 0, stream>>>(scores, mask, enc, attn_out,
                                                    fcin);

  k_gru<<<dim3(kB), dim3(256), 0, stream>>>(fcin, hidden, W_ih, W_hh, b_ih,
                                            b_hh, out_h);

  // prediction = fcin @ fc_W + fc_b   (M=128,N=32000,K=1536)
  k_gemm_bias<<<dim3(kV / 256, kB / 16), dim3(256), 0, stream>>>(
      fcin, fc_W, fc_b, pred, kV, kFCK);
}